// LSTMCell_68745246539792
// MI455X (gfx1250) — compile-verified
//
#include <hip/hip_runtime.h>
#include <math.h>

// Problem constants (match reference)
#define T_STEPS 2048
#define BATCH   32
#define INDIM   512
#define HID     512
#define GATES   2048          // 4*HID, gate order: f | i | g | o
#define KTOT    1024          // INDIM + HID

// Decomposition
#define NWG     64            // persistent workgroups
#define HCOLS   8             // h-columns per WG  (64*8 = 512)
#define LCOLS   32            // gate columns per WG (4 quadrants x 8)
#define RPITCH  520           // padded LDS row pitch (floats): 2x(256+4) DWORDs

typedef float    v2f __attribute__((ext_vector_type(2)));
typedef float    v8f __attribute__((ext_vector_type(8)));
typedef unsigned v4u __attribute__((ext_vector_type(4)));
typedef int      v4i __attribute__((ext_vector_type(4)));
typedef int      v8i __attribute__((ext_vector_type(8)));

// A-operand LDS layout (matches TDM pad: +4 DW after every 256 DW):
//   elem(r, c) at float offset  r*520 + c + 4*(c >= 256)

#if __has_builtin(__builtin_amdgcn_tensor_load_to_lds) && \
    __has_builtin(__builtin_amdgcn_s_wait_tensorcnt)
#define HAVE_TDM 1

typedef __attribute__((address_space(3))) void lds_void;
static __device__ inline unsigned lds_offset(const void* p) {
    return (unsigned)(uintptr_t)(lds_void*)p;
}

// One 2-D TDM tile load: [32 rows x 512 f32], row stride 512 in memory,
// LDS pitch 520 floats via pad_interval=256DW (code 7), pad_amount=4DW (code 3).
static __device__ inline void tdm_load_tile(unsigned lds_off, const void* gsrc) {
    const uintptr_t ga = (uintptr_t)gsrc;
    v4u g0;
    g0.x = 1u;                                   // count=1, user mode
    g0.y = lds_off;                              // lds_addr (bytes)
    g0.z = (unsigned)(ga & 0xFFFFFFFFu);         // global_addr[31:0]
    g0.w = (unsigned)((ga >> 32) & 0x01FFFFFFu)  // global_addr[56:32]
         | (2u << 30);                           // type = 2 ("image")
    v8i g1;
    g1[0] = (int)((2u << 16)      // data_size = 4 bytes
                | (1u << 20)      // pad_enable
                | (7u << 22)      // pad_interval = 256 DWORDs
                | (3u << 25));    // pad_amount  = 4 DWORDs
    g1[1] = (int)(512u << 16);    // tensor_dim0 = 512 (low 16 in [31:16])
    g1[2] = (int)(32u  << 16);    // tensor_dim0 hi=0 | tensor_dim1 = 32
    g1[3] = (int)(512u << 16);    // tensor_dim1 hi=0 | tile_dim0  = 512
    g1[4] = (int)(32u);           // tile_dim1 = 32, tile_dim2 = 0
    g1[5] = (int)(512);           // tensor_dim0_stride[31:0] = 512
    g1[6] = 0;
    g1[7] = 0;
    v4i z = {0, 0, 0, 0};
#if __has_include(<hip/amd_detail/amd_gfx1250_TDM.h>)
    v8i z8 = {0, 0, 0, 0, 0, 0, 0, 0};
    __builtin_amdgcn_tensor_load_to_lds(g0, g1, z, z, z8, 0);   // clang-23 form
#else
    __builtin_amdgcn_tensor_load_to_lds(g0, g1, z, z, 0);       // ROCm 7.2 form
#endif
}
#endif // HAVE_TDM

// One-shot weight repack: W[1024][2048] -> Wp[k4][col][4] so each lane's
// B-fragment (rows k4*4+koff, +koff+1 at one column) is a contiguous 8B load.
__global__ void __launch_bounds__(256)
pack_weights(const float* __restrict__ w, float* __restrict__ wp) {
    const int idx = blockIdx.x * 256 + threadIdx.x;   // 0 .. 1024*2048-1
    const int k   = idx >> 11;
    const int col = idx & 2047;
    const int k4  = k >> 2;
    const int r   = k & 3;                            // dest sub-slot == r
    wp[(size_t)k4 * (GATES * 4) + (size_t)col * 4 + r] = w[idx];
}

template <bool PACKED>
__global__ void __launch_bounds__(256)
lstm_persistent(const float* __restrict__ x,     // [T,B,512]
                const float* __restrict__ h0,    // [1,B,512]
                const float* __restrict__ c0,    // [1,B,512]
                const float* __restrict__ w,     // [1024,2048] row-major
                const float* __restrict__ wp,    // packed weights (or null)
                const float* __restrict__ bias,  // [2048]
                float* __restrict__ out,         // T*B*H outs, then h_f, c_f
                unsigned* __restrict__ bar)      // grid barrier counter (zeroed)
{
    extern __shared__ float smem[];
    float* sX    = smem;                   // [32][RPITCH]  x_t staged
    float* sH    = smem + 32 * RPITCH;     // [32][RPITCH]  h_{t-1} staged
    float* sPart = smem + 64 * RPITCH;     // [2][32][32]   K-split partial gates
    float* sC    = sPart + 2 * 32 * LCOLS; // [256]         persistent cell slice

    const int tid  = threadIdx.x;
    const int wg   = blockIdx.x;           // 0..63
    const int lane = tid & 31;
    const int wv   = tid >> 5;             // wave 0..7
    const int kw   = wv & 1;               // 0: x-part (K 0..511), 1: h-part
    const int mi   = (wv >> 1) & 1;        // M tile (rows 0..15 / 16..31)
    const int nj   = wv >> 2;              // N tile (local cols 0..15 / 16..31)

    // ---- WMMA per-lane constants (fp32 16x16x4 layout) ----
    const int a_m  = mi * 16 + (lane & 15);
    const int koff = (lane >> 4) << 1;     // 0 or 2 (K pair select)
    const int bn   = lane & 15;
    const int lcol = nj * 16 + bn;                                  // 0..31
    const int gcol = ((lcol >> 3) << 9) + wg * HCOLS + (lcol & 7);  // q*512+wg*8+j
    const float* bBase = PACKED
        ? wp + (size_t)(kw * 128) * (GATES * 4) + (size_t)gcol * 4 + koff
        : w  + (size_t)(kw * 512 + koff) * GATES + gcol;
    const float* aBase = (kw ? sH : sX) + a_m * RPITCH + koff;

    // ---- update-phase constants ----
    const int m_u = tid >> 3;
    const int j_u = tid & 7;
    const int hcol_u = wg * HCOLS + j_u;

    sC[tid] = c0[(size_t)m_u * HID + hcol_u];

#ifdef HAVE_TDM
    const unsigned sX_off = lds_offset(sX);
    const unsigned sH_off = lds_offset(sH);
#endif

    unsigned target = 0;
    for (int t = 0; t < T_STEPS; ++t) {
        const float* hsrc = (t == 0) ? h0 : (out + (size_t)(t - 1) * BATCH * HID);
#ifdef HAVE_TDM
        // ---- stage x_t and h_{t-1} via Tensor Data Mover (wave 0 only) ----
        if (wv == 0) {
            tdm_load_tile(sX_off, x + (size_t)t * BATCH * INDIM);
            tdm_load_tile(sH_off, hsrc);
            __builtin_amdgcn_s_wait_tensorcnt(0);
        }
        __syncthreads();
#else
        // ---- fallback: cooperative b128 staging (same padded layout) ----
        {
            const float4* xs = (const float4*)(x + (size_t)t * BATCH * INDIM);
            const float4* hs = (const float4*)hsrc;
            for (int i = tid; i < (BATCH * INDIM) / 4; i += 256) {
                const int row = i >> 7;
                const int col = (i & 127) << 2;
                const int off = row * RPITCH + col + ((col >= 256) ? 4 : 0);
                *(float4*)&sX[off] = xs[i];
                *(float4*)&sH[off] = hs[i];
            }
        }
        __syncthreads();
#endif

        // ---- K-split fp32 WMMA: each wave does 512 of K for one 16x16 tile ----
        v8f acc = {0.f, 0.f, 0.f, 0.f, 0.f, 0.f, 0.f, 0.f};
        const float* ap = aBase;
        const float* bp = bBase;
#pragma unroll 8
        for (int k = 0; k < 256; k += 4) {
            v2f a = *(const v2f*)ap;
            v2f b;
            if (PACKED) { b = *(const v2f*)bp; }
            else        { b.x = bp[0]; b.y = bp[GATES]; }
            acc = __builtin_amdgcn_wmma_f32_16x16x4_f32(
                      false, a, false, b, (short)0, acc, false, false);
            ap += 4;
            bp += PACKED ? (GATES * 4) : (size_t)4 * GATES;
        }
        ap += 4;  // skip mid-row LDS pad (4 DWORDs after 256)
#pragma unroll 8
        for (int k = 256; k < 512; k += 4) {
            v2f a = *(const v2f*)ap;
            v2f b;
            if (PACKED) { b = *(const v2f*)bp; }
            else        { b.x = bp[0]; b.y = bp[GATES]; }
            acc = __builtin_amdgcn_wmma_f32_16x16x4_f32(
                      false, a, false, b, (short)0, acc, false, false);
            ap += 4;
            bp += PACKED ? (GATES * 4) : (size_t)4 * GATES;
        }
        // scatter accumulator to LDS partial buffer
        {
            const int crow = mi * 16 + ((lane >> 4) << 3);
            float* pp = sPart + (size_t)(kw * 32 + crow) * LCOLS + lcol;
#pragma unroll
            for (int r = 0; r < 8; ++r) pp[r * LCOLS] = acc[r];
        }
        __syncthreads();

        // ---- gate fusion + state update: one thread per (batch row, h col) ----
        {
            float g[4];
#pragma unroll
            for (int q = 0; q < 4; ++q) {
                const int lc = q * HCOLS + j_u;
                g[q] = sPart[(size_t)m_u * LCOLS + lc]
                     + sPart[(size_t)(32 + m_u) * LCOLS + lc]
                     + bias[q * HID + hcol_u];
            }
            const float fg = 1.f / (1.f + expf(-g[0]));
            const float ig = 1.f / (1.f + expf(-g[1]));
            const float gg = tanhf(g[2]);
            const float og = 1.f / (1.f + expf(-g[3]));
            const float c  = fg * sC[tid] + ig * gg;
            sC[tid] = c;
            const float h = og * tanhf(c);
            out[(size_t)t * BATCH * HID + (size_t)m_u * HID + hcol_u] = h;
            if (t == T_STEPS - 1) {
                const size_t base = (size_t)T_STEPS * BATCH * HID;
                const size_t idx  = (size_t)m_u * HID + hcol_u;
                out[base + idx] = h;                      // h_f
                out[base + BATCH * HID + idx] = c;        // c_f
            }
        }

        // ---- grid barrier: release h_t, wait for all 64 WGs ----
        __threadfence();
        __syncthreads();
        target += NWG;
        if (tid == 0) {
            __hip_atomic_fetch_add(bar, 1u, __ATOMIC_RELEASE,
                                   __HIP_MEMORY_SCOPE_AGENT);
            while (__hip_atomic_load(bar, __ATOMIC_ACQUIRE,
                                     __HIP_MEMORY_SCOPE_AGENT) < target)
                __builtin_amdgcn_s_sleep(1);
        }
        __syncthreads();
        __threadfence();
    }
}

extern "C" void kernel_launch(void* const* d_in, const int* in_sizes, int n_in,
                              void* d_out, int out_size, void* d_ws, size_t ws_size,
                              hipStream_t stream) {
    (void)in_sizes; (void)n_in; (void)out_size;
    const float* x    = (const float*)d_in[0];
    const float* h0   = (const float*)d_in[1];
    const float* c0   = (const float*)d_in[2];
    const float* w    = (const float*)d_in[3];
    const float* bias = (const float*)d_in[4];
    float* out        = (float*)d_out;
    unsigned* bar     = (unsigned*)d_ws;

    // re-zero the grid-barrier counter every call (graph-capture safe)
    hipMemsetAsync(d_ws, 0, 64, stream);

    const size_t pack_off   = 256;
    const size_t pack_bytes = (size_t)KTOT * GATES * sizeof(float);  // 8 MB
    const bool   packed     = ws_size >= pack_off + pack_bytes;
    float* wpk = (float*)((char*)d_ws + pack_off);

    const size_t smem_bytes =
        (64 * RPITCH + 2 * 32 * LCOLS + 256) * sizeof(float);        // ~139 KB

    if (packed) {
        pack_weights<<<(KTOT * GATES) / 256, 256, 0, stream>>>(w, wpk);
        lstm_persistent<true><<<NWG, 256, smem_bytes, stream>>>(
            x, h0, c0, w, wpk, bias, out, bar);
    } else {
        lstm_persistent<false><<<NWG, 256, smem_bytes, stream>>>(
            x, h0, c0, w, nullptr, bias, out, bar);
    }
}